// MyModel_87522843558845
// MI455X (gfx1250) — compile-verified
//
#include <hip/hip_runtime.h>

typedef __attribute__((ext_vector_type(2))) float v2f;
typedef __attribute__((ext_vector_type(4))) float v4f;
typedef __attribute__((ext_vector_type(8))) float v8f;

#define ALPHA 0.002
#define NITER 100

// ---------------------------------------------------------------------------
// Preamble: compute P = M^100 and S = sum_{t=0..99} M^t  (M = I - alpha*A),
// in double precision, then emit W[16][16] (row-major float):
//   W[m][k] = (m<8) ? ( k<8 ? P[k][m] : alpha*S[k-8][m] ) : 0
// W is the constant WMMA A-operand:  out^T(16x16 tile) = W * [x^T ; y^T].
// One block of 256 threads; threads >= 64 only help write W.
// ---------------------------------------------------------------------------
__global__ void prep_W(const float* __restrict__ A, float* __restrict__ W) {
    __shared__ double Ms[64], Ps[64], Ss[64];
    const int t = threadIdx.x;
    if (t < 64) {
        const int i = t >> 3, j = t & 7;
        Ms[t] = ((i == j) ? 1.0 : 0.0) - ALPHA * (double)A[t];
        Ps[t] = (i == j) ? 1.0 : 0.0;
        Ss[t] = 0.0;
    }
    __syncthreads();
    for (int it = 0; it < NITER; ++it) {
        double p = 0.0;
        if (t < 64) {
            const int i = t >> 3, j = t & 7;
            Ss[t] += Ps[t];                       // S accumulates M^it
            for (int k = 0; k < 8; ++k)
                p += Ps[i * 8 + k] * Ms[k * 8 + j];
        }
        __syncthreads();
        if (t < 64) Ps[t] = p;                   // P becomes M^(it+1)
        __syncthreads();
    }
    const int m = t >> 4, k = t & 15;
    float w = 0.0f;
    if (m < 8)
        w = (k < 8) ? (float)Ps[k * 8 + m]
                    : (float)(ALPHA * Ss[(k - 8) * 8 + m]);
    W[t] = w;
}

// ---------------------------------------------------------------------------
// Main pass: one wave per 16-row tile.
//   out^T = W(16x16) * Z(16x16),  Z = [x^T ; y^T]
// computed as 4 chained V_WMMA_F32_16X16X4_F32 (K = 16 split into 4 chunks).
// Per-lane operand layouts follow ISA 7.12.2 (32-bit A 16x4 / B 4x16).
// ---------------------------------------------------------------------------
__global__ void fused_scan_wmma(const float* __restrict__ x,
                                const float* __restrict__ y,
                                const float* __restrict__ W,
                                float* __restrict__ out,
                                int numTiles) {
    const int wave = threadIdx.x >> 5;
    const int lane = threadIdx.x & 31;
    const long tile = (long)blockIdx.x * (blockDim.x >> 5) + wave;
    if (tile >= numTiles) return;              // wave-uniform: EXEC stays full

    const int r  = lane & 15;                  // A: m index / B: column n (tile row)
    const int hi = lane >> 4;                  // lane half selects K pair

    // A operand (constant W): lane holds W[m, 4c + 2*hi + {0,1}]
    const float* wrow = W + r * 16 + 2 * hi;
    const v2f a0 = *(const v2f*)(wrow + 0);
    const v2f a1 = *(const v2f*)(wrow + 4);
    const v2f a2 = *(const v2f*)(wrow + 8);
    const v2f a3 = *(const v2f*)(wrow + 12);

    // B operand (Z): rows 0..7 = x^T, rows 8..15 = y^T.
    // Lane holds Z[4c + 2*hi + {0,1}, r] = (x|y)[tile*16 + r, consecutive cols]
    const float* xrow = x + (tile * 16 + r) * 8 + 2 * hi;
    const float* yrow = y + (tile * 16 + r) * 8 + 2 * hi;
    const v2f b0 = *(const v2f*)(xrow + 0);
    const v2f b1 = *(const v2f*)(xrow + 4);
    const v2f b2 = *(const v2f*)(yrow + 0);
    const v2f b3 = *(const v2f*)(yrow + 4);

    v8f d = {};
    d = __builtin_amdgcn_wmma_f32_16x16x4_f32(false, a0, false, b0, (short)0, d, false, false);
    d = __builtin_amdgcn_wmma_f32_16x16x4_f32(false, a1, false, b1, (short)0, d, false, false);
    d = __builtin_amdgcn_wmma_f32_16x16x4_f32(false, a2, false, b2, (short)0, d, false, false);
    d = __builtin_amdgcn_wmma_f32_16x16x4_f32(false, a3, false, b3, (short)0, d, false, false);

    // D layout: lanes 0..15, VGPR v holds out[tile*16 + lane, v] (v = 0..7).
    // Lanes 16..31 hold the zero-pad rows of W -> discard.
    if (lane < 16) {
        float* o = out + (tile * 16 + lane) * 8;
        v4f lo4, hi4;
        lo4.x = d[0]; lo4.y = d[1]; lo4.z = d[2]; lo4.w = d[3];
        hi4.x = d[4]; hi4.y = d[5]; hi4.z = d[6]; hi4.w = d[7];
        *(v4f*)(o)     = lo4;
        *(v4f*)(o + 4) = hi4;
    }
}

// Scalar fallback for rows % 16 (zero for BATCH = 2M, kept for safety).
__global__ void tail_rows(const float* __restrict__ x,
                          const float* __restrict__ y,
                          const float* __restrict__ W,
                          float* __restrict__ out,
                          long startRow, long nRows) {
    const long i = (long)blockIdx.x * blockDim.x + threadIdx.x;
    if (i >= nRows) return;
    const long r = startRow + i;
    float xv[8], yv[8];
    for (int j = 0; j < 8; ++j) { xv[j] = x[r * 8 + j]; yv[j] = y[r * 8 + j]; }
    for (int c = 0; c < 8; ++c) {
        float acc = 0.0f;
        for (int j = 0; j < 8; ++j)
            acc += W[c * 16 + j] * xv[j] + W[c * 16 + 8 + j] * yv[j];
        out[r * 8 + c] = acc;
    }
}

extern "C" void kernel_launch(void* const* d_in, const int* in_sizes, int n_in,
                              void* d_out, int out_size, void* d_ws, size_t ws_size,
                              hipStream_t stream) {
    const float* x = (const float*)d_in[0];
    const float* y = (const float*)d_in[1];
    const float* A = (const float*)d_in[2];
    float* out = (float*)d_out;
    float* W   = (float*)d_ws;               // 16*16 floats = 1 KB scratch

    const long rows  = (long)in_sizes[0] / 8;
    const long tiles = rows / 16;
    const long rem   = rows - tiles * 16;

    prep_W<<<1, 256, 0, stream>>>(A, W);

    if (tiles > 0) {
        const int wavesPerBlock = 8;        // 256 threads = 8 wave32
        const long blocks = (tiles + wavesPerBlock - 1) / wavesPerBlock;
        fused_scan_wmma<<<(int)blocks, 32 * wavesPerBlock, 0, stream>>>(
            x, y, W, out, (int)tiles);
    }
    if (rem > 0) {
        tail_rows<<<(int)((rem + 255) / 256), 256, 0, stream>>>(
            x, y, W, out, tiles * 16, rem);
    }
}